// SE_Block_54468775248341
// MI455X (gfx1250) — compile-verified
//
#include <hip/hip_runtime.h>
#include <hip/hip_bf16.h>
#include <math.h>

// ---------------------------------------------------------------------------
// SE block for MI455X (gfx1250, wave32).
// Memory-bound: 205.5 MB pooled read + 411 MB scale read/write ~= 26 us floor
// at 23.3 TB/s. GEMMs are tiny -> done on one WGP with fp32 WMMA
// (v_wmma_f32_16x16x4_f32) for exact precision parity with the reference.
// ---------------------------------------------------------------------------

#define B_ 32
#define C_ 512
#define HID_ 32
#define HW_ 3136        // 56*56
#define HW4_ 784        // HW_/4
#define POS_IDX 409     // round(0.8*512) - 1

typedef float v2f __attribute__((ext_vector_type(2)));
typedef float v8f __attribute__((ext_vector_type(8)));

// ---------------------------------------------------------------------------
// Kernel 1: global average pool. One block per (b,c) plane; 128-bit loads.
// ---------------------------------------------------------------------------
__global__ __launch_bounds__(256) void se_pool(const float* __restrict__ x,
                                               float* __restrict__ pooled) {
    const int plane = blockIdx.x;                       // b*C_ + c
    const float4* p = (const float4*)(x + (size_t)plane * HW_);
    float sum = 0.0f;
    for (int i = threadIdx.x; i < HW4_; i += 256) {
        float4 v = p[i];
        sum += (v.x + v.y) + (v.z + v.w);
    }
    // wave32 reduction
    for (int off = 16; off > 0; off >>= 1)
        sum += __shfl_down(sum, off, 32);
    __shared__ float ws[8];
    if ((threadIdx.x & 31) == 0) ws[threadIdx.x >> 5] = sum;
    __syncthreads();
    if (threadIdx.x == 0) {
        float t = 0.0f;
        #pragma unroll
        for (int i = 0; i < 8; ++i) t += ws[i];
        pooled[plane] = t * (1.0f / (float)HW_);
    }
}

// ---------------------------------------------------------------------------
// Kernel 2: z = relu(pooled @ W1^T); s = sigmoid(z @ W2^T).
// Single workgroup, 8 waves, fp32 WMMA 16x16x4.
// Fragment layouts per CDNA5 ISA 7.12.2:
//   A(16x4):  lane l, half h -> {A[l][k0+2h], A[l][k0+2h+1]}
//   B(4x16):  lane l, half h -> {B[k0+2h][n0+l], B[k0+2h+1][n0+l]}
//             with B = W^T this is {W[n0+l][k0+2h], W[n0+l][k0+2h+1]}
//   C(16x16): vgpr j, lane l, half h -> C[j + 8h][l]
// ---------------------------------------------------------------------------
__global__ __launch_bounds__(256) void se_wmma(const float* __restrict__ pooled,
                                               const float* __restrict__ W1,
                                               const float* __restrict__ W2,
                                               float* __restrict__ s) {
    __shared__ float zbuf[HID_][HID_ + 1];   // 32x33, padded vs bank conflicts

    const int wid  = threadIdx.x >> 5;       // wave id 0..7
    const int lane = threadIdx.x & 31;
    const int h    = lane >> 4;              // half: 0 or 1
    const int l    = lane & 15;

    // ---- GEMM1: M=32(batch) N=32(hid) K=512, 4 C-tiles on waves 0..3 ----
    if (wid < 4) {
        const int mi = (wid & 1) * 16;       // batch-row tile base
        const int ni = (wid >> 1) * 16;      // hid-col tile base
        v8f acc = {};
        for (int k0 = 0; k0 < C_; k0 += 4) {
            v2f a, b;
            a.x = pooled[(mi + l) * C_ + k0 + 2 * h];
            a.y = pooled[(mi + l) * C_ + k0 + 2 * h + 1];
            b.x = W1[(ni + l) * C_ + k0 + 2 * h];
            b.y = W1[(ni + l) * C_ + k0 + 2 * h + 1];
            acc = __builtin_amdgcn_wmma_f32_16x16x4_f32(
                false, a, false, b, (short)0, acc, false, false);
        }
        #pragma unroll
        for (int j = 0; j < 8; ++j) {
            float v = acc[j];
            zbuf[mi + j + 8 * h][ni + l] = v > 0.0f ? v : 0.0f;  // ReLU
        }
    }
    __syncthreads();

    // ---- GEMM2: M=32(batch) N=512(chan) K=32(hid), 64 C-tiles on 8 waves ----
    for (int t = wid; t < 64; t += 8) {
        const int mi = (t & 1) * 16;
        const int ni = (t >> 1) * 16;
        v8f acc = {};
        #pragma unroll
        for (int k0 = 0; k0 < HID_; k0 += 4) {
            v2f a, b;
            a.x = zbuf[mi + l][k0 + 2 * h];
            a.y = zbuf[mi + l][k0 + 2 * h + 1];
            b.x = W2[(ni + l) * HID_ + k0 + 2 * h];
            b.y = W2[(ni + l) * HID_ + k0 + 2 * h + 1];
            acc = __builtin_amdgcn_wmma_f32_16x16x4_f32(
                false, a, false, b, (short)0, acc, false, false);
        }
        #pragma unroll
        for (int j = 0; j < 8; ++j) {
            float v = acc[j];
            s[(mi + j + 8 * h) * C_ + ni + l] = 1.0f / (1.0f + __expf(-v));
        }
    }
}

// ---------------------------------------------------------------------------
// Kernel 3: per-batch-row rank selection -> mask. One block per row,
// 512 threads; stable brute-force rank (512x512 compares, fully parallel).
// mask[b][c] = (s[b][c] <= sorted_row[POS_IDX]) ? 1 : 0
// ---------------------------------------------------------------------------
__global__ __launch_bounds__(512) void se_select(const float* __restrict__ s,
                                                 float* __restrict__ mask) {
    __shared__ float row[C_];
    __shared__ float thr;
    const int b = blockIdx.x;
    const int j = threadIdx.x;
    row[j] = s[b * C_ + j];
    __syncthreads();
    const float v = row[j];
    int rank = 0;
    for (int i = 0; i < C_; ++i) {
        float u = row[i];
        rank += (u < v) || (u == v && i < j);   // stable rank
    }
    if (rank == POS_IDX) thr = v;               // exactly one writer
    __syncthreads();
    mask[b * C_ + j] = (v <= thr) ? 1.0f : 0.0f;
}

// ---------------------------------------------------------------------------
// Kernel 4: out = x * mask (broadcast over H,W). One block per plane,
// 128-bit load/store streaming pass.
// ---------------------------------------------------------------------------
__global__ __launch_bounds__(256) void se_scale(const float* __restrict__ x,
                                                const float* __restrict__ mask,
                                                float* __restrict__ out) {
    const int plane = blockIdx.x;               // b*C_ + c
    const float m = mask[plane];
    const float4* xi = (const float4*)(x + (size_t)plane * HW_);
    float4* yo = (float4*)(out + (size_t)plane * HW_);
    for (int i = threadIdx.x; i < HW4_; i += 256) {
        float4 v = xi[i];
        v.x *= m; v.y *= m; v.z *= m; v.w *= m;
        yo[i] = v;
    }
}

// ---------------------------------------------------------------------------
extern "C" void kernel_launch(void* const* d_in, const int* in_sizes, int n_in,
                              void* d_out, int out_size, void* d_ws, size_t ws_size,
                              hipStream_t stream) {
    const float* x  = (const float*)d_in[0];   // (32,512,56,56)
    const float* W1 = (const float*)d_in[1];   // (32,512)
    const float* W2 = (const float*)d_in[2];   // (512,32)
    float* out = (float*)d_out;                // (32,512,56,56)

    float* pooled = (float*)d_ws;              // 32*512 floats
    float* s      = pooled + B_ * C_;          // 32*512 floats
    float* mask   = s + B_ * C_;               // 32*512 floats

    se_pool  <<<B_ * C_, 256, 0, stream>>>(x, pooled);
    se_wmma  <<<1,       256, 0, stream>>>(pooled, W1, W2, s);
    se_select<<<B_,      512, 0, stream>>>(s, mask);
    se_scale <<<B_ * C_, 256, 0, stream>>>(x, mask, out);
}